// ProductSpaceMessagePassing_47596827574580
// MI455X (gfx1250) — compile-verified
//
#include <hip/hip_runtime.h>

#define FEAT 64

typedef float v2f __attribute__((ext_vector_type(2)));
typedef float v8f __attribute__((ext_vector_type(8)));

__device__ __forceinline__ float wave_sum(float v) {
#pragma unroll
  for (int m = 16; m >= 1; m >>= 1) v += __shfl_xor(v, m, 32);
  return v;
}

// ---------------------------------------------------------------------------
// H[n x 64] = X[n x 64] @ W^T + bias   (W row-major 64x64, H[i,j]=sum_k X[i,k]W[j,k])
// One wave handles a 16-row tile; full N=64 via 4 accumulators; K=64 in steps of 4
// using V_WMMA_F32_16X16X4_F32.
// A layout: lane L -> M = L&15, K = 2*(L>>4) + {0,1}
// B layout: lane L -> N = L&15, K = 2*(L>>4) + {0,1}; B[k][n] = W[n][k]
// C/D layout: VGPR r -> (lanes 0-15: M=r, lanes 16-31: M=r+8), N = lane&15
// ---------------------------------------------------------------------------
__global__ __launch_bounds__(256) void gemm64_wmma(const float* __restrict__ X,
                                                   const float* __restrict__ W,
                                                   const float* __restrict__ bias,
                                                   float* __restrict__ H, int n) {
  __shared__ float lw[64 * 66];  // W[j][k] at lw[j*66+k]; stride 66 -> bank-conflict free, 8B aligned
  const int tid = threadIdx.x;
  for (int i = tid; i < 64 * 64; i += 256) lw[(i >> 6) * 66 + (i & 63)] = W[i];
  __syncthreads();

  const int wave = tid >> 5;
  const int lane = tid & 31;
  const int tile = blockIdx.x * 8 + wave;
  const int ntiles = (n + 15) >> 4;
  if (tile >= ntiles) return;

  const int row_base = tile * 16;
  const int m = lane & 15;
  const int khalf = (lane >> 4) * 2;  // 0 or 2
  const int arow = min(row_base + m, n - 1);
  const float* xrow = X + (size_t)arow * FEAT;

  v8f acc0 = {}, acc1 = {}, acc2 = {}, acc3 = {};
#pragma unroll
  for (int k0 = 0; k0 < 64; k0 += 4) {
    v2f a = *(const v2f*)(xrow + k0 + khalf);
    v2f b0 = *(const v2f*)(&lw[(m) * 66 + k0 + khalf]);
    v2f b1 = *(const v2f*)(&lw[(16 + m) * 66 + k0 + khalf]);
    v2f b2 = *(const v2f*)(&lw[(32 + m) * 66 + k0 + khalf]);
    v2f b3 = *(const v2f*)(&lw[(48 + m) * 66 + k0 + khalf]);
    acc0 = __builtin_amdgcn_wmma_f32_16x16x4_f32(false, a, false, b0, (short)0, acc0, false, false);
    acc1 = __builtin_amdgcn_wmma_f32_16x16x4_f32(false, a, false, b1, (short)0, acc1, false, false);
    acc2 = __builtin_amdgcn_wmma_f32_16x16x4_f32(false, a, false, b2, (short)0, acc2, false, false);
    acc3 = __builtin_amdgcn_wmma_f32_16x16x4_f32(false, a, false, b3, (short)0, acc3, false, false);
  }

  const int rowsel = (lane < 16) ? 0 : 8;
  const float bi0 = bias[m], bi1 = bias[16 + m], bi2 = bias[32 + m], bi3 = bias[48 + m];
#pragma unroll
  for (int r = 0; r < 8; ++r) {
    int row = row_base + rowsel + r;
    if (row >= n) continue;
    float* hrow = H + (size_t)row * FEAT;
    hrow[m]      = acc0[r] + bi0;
    hrow[16 + m] = acc1[r] + bi1;
    hrow[32 + m] = acc2[r] + bi2;
    hrow[48 + m] = acc3[r] + bi3;
  }
}

// ---------------------------------------------------------------------------
// Pre-maps (wave32 per node, 2 elems/lane):
//  tb = log_map0(b) = (2/sqrt(c)) * atanh(sqrt(c)*||b||)/||b|| * b
//  sn = s / max(||s||, 1e-12)
// In-place safe (each lane writes only what it read).
// ---------------------------------------------------------------------------
__global__ __launch_bounds__(256) void premap_kernel(const float* __restrict__ b_in,
                                                     float* __restrict__ tb_out,
                                                     const float* __restrict__ s_in,
                                                     float* __restrict__ sn_out,
                                                     const float* __restrict__ c_ptr, int n) {
  int node = (int)((blockIdx.x * blockDim.x + threadIdx.x) >> 5);
  int lane = threadIdx.x & 31;
  if (node >= n) return;
  const float sc = sqrtf(c_ptr[0]);
  size_t base = (size_t)node * FEAT + lane * 2;
  float b0 = b_in[base], b1 = b_in[base + 1];
  float s0 = s_in[base], s1 = s_in[base + 1];
  float rb = sqrtf(wave_sum(b0 * b0 + b1 * b1));
  float rs = sqrtf(wave_sum(s0 * s0 + s1 * s1));
  float arg = fminf(sc * rb, 0.999999940395f);
  float scale_b = (rb > 1e-20f) ? (2.0f / sc) * atanhf(arg) / rb : 2.0f;
  float scale_s = 1.0f / fmaxf(rs, 1e-12f);
  tb_out[base] = b0 * scale_b; tb_out[base + 1] = b1 * scale_b;
  sn_out[base] = s0 * scale_s; sn_out[base + 1] = s1 * scale_s;
}

// L2-normalize rows in place (inner spherical normalize after GEMM)
__global__ __launch_bounds__(256) void rownorm_kernel(float* __restrict__ X, int n) {
  int node = (int)((blockIdx.x * blockDim.x + threadIdx.x) >> 5);
  int lane = threadIdx.x & 31;
  if (node >= n) return;
  size_t base = (size_t)node * FEAT + lane * 2;
  float x0 = X[base], x1 = X[base + 1];
  float nn = sqrtf(wave_sum(x0 * x0 + x1 * x1));
  float s = 1.0f / fmaxf(nn, 1e-12f);
  X[base] = x0 * s; X[base + 1] = x1 * s;
}

__global__ __launch_bounds__(256) void zero_f(float* __restrict__ p, long long n) {
  long long i = (long long)blockIdx.x * blockDim.x + threadIdx.x;
  if (i < n) p[i] = 0.0f;
}

__global__ __launch_bounds__(256) void deg_count(const int* __restrict__ dst,
                                                 unsigned* __restrict__ deg, int e) {
  int i = blockIdx.x * blockDim.x + threadIdx.x;
  if (i < e) atomicAdd(&deg[dst[i]], 1u);
}

__global__ __launch_bounds__(256) void deg_invert(const unsigned* __restrict__ deg,
                                                  float* __restrict__ inv, int n) {
  int i = blockIdx.x * blockDim.x + threadIdx.x;
  if (i < n) inv[i] = 1.0f / fmaxf((float)deg[i], 1.0f);
}

// Scatter-sum over edges, all 3 spaces fused. Thread = (edge, float4 chunk).
__global__ __launch_bounds__(256) void scatter_kernel(
    const float4* __restrict__ he, const float4* __restrict__ hb, const float4* __restrict__ hs,
    float* __restrict__ ae, float* __restrict__ ab, float* __restrict__ as_,
    const int* __restrict__ src, const int* __restrict__ dst, int ne) {
  long long idx = (long long)blockIdx.x * blockDim.x + threadIdx.x;
  if (idx >= (long long)ne * 16) return;
  int e = (int)(idx >> 4);
  int q = (int)(idx & 15);
  int s = src[e], d = dst[e];
  size_t sq = (size_t)s * 16 + q;
  size_t doff = (size_t)d * FEAT + q * 4;
  float4 v = he[sq];
  unsafeAtomicAdd(ae + doff + 0, v.x); unsafeAtomicAdd(ae + doff + 1, v.y);
  unsafeAtomicAdd(ae + doff + 2, v.z); unsafeAtomicAdd(ae + doff + 3, v.w);
  v = hb[sq];
  unsafeAtomicAdd(ab + doff + 0, v.x); unsafeAtomicAdd(ab + doff + 1, v.y);
  unsafeAtomicAdd(ab + doff + 2, v.z); unsafeAtomicAdd(ab + doff + 3, v.w);
  v = hs[sq];
  unsafeAtomicAdd(as_ + doff + 0, v.x); unsafeAtomicAdd(as_ + doff + 1, v.y);
  unsafeAtomicAdd(as_ + doff + 2, v.z); unsafeAtomicAdd(as_ + doff + 3, v.w);
}

// Post-aggregation: mean, then leaky_relu / exp_map0 / l2-normalize.
__global__ __launch_bounds__(256) void post_kernel(
    const float* __restrict__ ae, const float* __restrict__ ab, const float* __restrict__ as_,
    const float* __restrict__ inv_deg, const float* __restrict__ c_ptr,
    float* __restrict__ eo, float* __restrict__ bo, float* __restrict__ so, int n) {
  int node = (int)((blockIdx.x * blockDim.x + threadIdx.x) >> 5);
  int lane = threadIdx.x & 31;
  if (node >= n) return;
  float w = inv_deg[node];
  size_t base = (size_t)node * FEAT + lane * 2;
  float e0 = ae[base] * w, e1 = ae[base + 1] * w;
  eo[base]     = e0 > 0.f ? e0 : 0.2f * e0;
  eo[base + 1] = e1 > 0.f ? e1 : 0.2f * e1;
  float v0 = ab[base] * w, v1 = ab[base + 1] * w;
  float u0 = as_[base] * w, u1 = as_[base + 1] * w;
  float vn = sqrtf(wave_sum(v0 * v0 + v1 * v1));
  float un = sqrtf(wave_sum(u0 * u0 + u1 * u1));
  float sc = sqrtf(c_ptr[0]);
  float bscale = (vn > 1e-20f) ? tanhf(0.5f * sc * vn) / (sc * vn) : 0.5f;
  bo[base] = v0 * bscale; bo[base + 1] = v1 * bscale;
  float sscale = 1.0f / fmaxf(un, 1e-12f);
  so[base] = u0 * sscale; so[base + 1] = u1 * sscale;
}

extern "C" void kernel_launch(void* const* d_in, const int* in_sizes, int n_in,
                              void* d_out, int out_size, void* d_ws, size_t ws_size,
                              hipStream_t stream) {
  const float* e_emb = (const float*)d_in[0];
  const float* b_emb = (const float*)d_in[1];
  const float* s_emb = (const float*)d_in[2];
  const float* b_c   = (const float*)d_in[3];
  // d_in[4] = s_curvature: unused by the reference math
  const float* e_W = (const float*)d_in[5];
  const float* e_b = (const float*)d_in[6];
  const float* b_W = (const float*)d_in[7];
  const float* b_b = (const float*)d_in[8];
  const float* s_W = (const float*)d_in[9];
  const float* s_b = (const float*)d_in[10];
  const int* src = (const int*)d_in[11];
  const int* dst = (const int*)d_in[12];

  const int N = in_sizes[0] / FEAT;
  const int E = in_sizes[11];
  const size_t nd = (size_t)N * FEAT;

  float* out   = (float*)d_out;
  float* out_e = out;
  float* out_b = out + nd;
  float* out_s = out + 2 * nd;

  float* ws = (float*)d_ws;
  float* h_e = ws;
  float* h_b = ws + nd;
  float* h_s = ws + 2 * nd;
  float* agg = ws + 3 * nd;            // 3*nd contiguous (agg_e|agg_b|agg_s)
  float* agg_e = agg;
  float* agg_b = agg + nd;
  float* agg_s = agg + 2 * nd;
  unsigned* deg  = (unsigned*)(ws + 6 * nd);
  float* inv_deg = ws + 6 * nd + N;

  const int TB = 256;
  const int ntiles = (N + 15) >> 4;
  const int gemm_blocks = (ntiles + 7) / 8;
  const int node_wave_blocks = (N + 7) / 8;       // 8 waves/block, wave per node
  const long long agg_elems = 3LL * (long long)nd;

  // degree (shared across layers/spaces)
  zero_f<<<(int)((N + TB - 1) / TB), TB, 0, stream>>>((float*)deg, N);
  deg_count<<<(E + TB - 1) / TB, TB, 0, stream>>>(dst, deg, E);
  deg_invert<<<(N + TB - 1) / TB, TB, 0, stream>>>(deg, inv_deg, N);

  for (int l = 0; l < 2; ++l) {
    const float* inE = (l == 0) ? e_emb : out_e;
    const float* inB = (l == 0) ? b_emb : out_b;
    const float* inS = (l == 0) ? s_emb : out_s;

    // pre-maps write into d_out regions (layer 1: in place)
    premap_kernel<<<node_wave_blocks, TB, 0, stream>>>(inB, out_b, inS, out_s, b_c, N);

    gemm64_wmma<<<gemm_blocks, TB, 0, stream>>>(inE,   e_W + (size_t)l * FEAT * FEAT, e_b + (size_t)l * FEAT, h_e, N);
    gemm64_wmma<<<gemm_blocks, TB, 0, stream>>>(out_b, b_W + (size_t)l * FEAT * FEAT, b_b + (size_t)l * FEAT, h_b, N);
    gemm64_wmma<<<gemm_blocks, TB, 0, stream>>>(out_s, s_W + (size_t)l * FEAT * FEAT, s_b + (size_t)l * FEAT, h_s, N);

    rownorm_kernel<<<node_wave_blocks, TB, 0, stream>>>(h_s, N);

    zero_f<<<(int)((agg_elems + TB - 1) / TB), TB, 0, stream>>>(agg, agg_elems);
    long long sc_threads = (long long)E * 16;
    scatter_kernel<<<(int)((sc_threads + TB - 1) / TB), TB, 0, stream>>>(
        (const float4*)h_e, (const float4*)h_b, (const float4*)h_s,
        agg_e, agg_b, agg_s, src, dst, E);

    post_kernel<<<node_wave_blocks, TB, 0, stream>>>(agg_e, agg_b, agg_s, inv_deg, b_c,
                                                     out_e, out_b, out_s, N);
  }
}